// WaveletTransform_13932873908737
// MI455X (gfx1250) — compile-verified
//
#include <hip/hip_runtime.h>

typedef __attribute__((ext_vector_type(2))) float v2f;
typedef __attribute__((ext_vector_type(4))) float v4f;
typedef __attribute__((ext_vector_type(8))) float v8f;

// wflip[f][t] = DEC_f[7-t]  (the correlation filters the reference applies to ext)
__constant__ float WFLIP[16] = {
    // lo reversed
    0.23037781330885523f,  0.7148465705525415f,  0.6308807679295904f,  -0.02798376941698385f,
    -0.18703481171888114f, 0.030841381835986965f, 0.032883011666982945f, -0.010597401784997278f,
    // hi reversed
    -0.010597401784997278f, -0.032883011666982945f, 0.030841381835986965f, 0.18703481171888114f,
    -0.02798376941698385f, -0.6308807679295904f,  0.7148465705525415f,  -0.23037781330885523f
};

// symmetric (half-sample) reflection of virtual index v into [0, N)
__device__ __forceinline__ int symref(int v, int N) {
    if (v < 0)  v = -1 - v;
    if (v >= N) v = 2 * N - 1 - v;
    if (v < 0)  v = 0;
    if (v >= N) v = N - 1;
    return v;
}

// ---------------------------------------------------------------------------
// Kernel 1: streaming zero-fill of the 262 MB output (non-temporal b128 stores)
// ---------------------------------------------------------------------------
__global__ __launch_bounds__(256) void fill_zero_kernel(float* __restrict__ out, long n) {
    long i = ((long)blockIdx.x * blockDim.x + threadIdx.x) * 4;
    if (i + 3 < n) {
        v4f z = {};
        __builtin_nontemporal_store(z, (v4f*)(out + i));
    } else {
        for (long j = i; j < n; ++j) out[j] = 0.0f;
    }
}

// ---------------------------------------------------------------------------
// Kernel 2: 5-level wavelet packet, one block per batch, all levels in LDS.
// Each wave-task computes 128 output positions x 2 filters of one channel via
// a chain of 6 accumulating v_wmma_f32_16x16x4_f32 (full 16x16 utilization
// using the shifted-filter-bank A matrix).
// ---------------------------------------------------------------------------
#define BUF_STRIDE 16200   // >= max level payload (32*506 = 16192 floats)

__global__ __launch_bounds__(256) void wavelet_wmma_kernel(const float* __restrict__ x,
                                                           float* __restrict__ out,
                                                           int tot) {
    __shared__ float lds[2 * BUF_STRIDE];   // 129,600 B: verifies gfx1250 big-LDS at compile time

    const int tid  = threadIdx.x;
    const int b    = blockIdx.x;
    const int lane = tid & 31;
    const int wave = tid >> 5;
    const int half = lane >> 4;   // lane half selects K-offset pair in A/B and M-offset in D
    const int mrow = lane & 15;   // A-matrix row (M) for this lane; also the D column index n
    const int f    = mrow >> 3;   // filter (0=lo, 1=hi) encoded in A row
    const int s    = mrow & 7;    // shift encoded in A row

    // ---- load input batch into LDS (4000 b128 transactions) ----
    {
        const v4f* src = (const v4f*)(x + (long)b * 16000);
        v4f* dst = (v4f*)lds;
        for (int i = tid; i < 4000; i += 256) dst[i] = src[i];
    }

    // ---- constant A fragments: A[f*8+s][k] = wflip_f[k - 2s], k in 0..23 (6 k-steps of 4) ----
    // f32 16x4 A layout: lanes0-15 hold M=lane, V0=K0,V1=K1; lanes16-31 hold M=lane-16, V0=K2,V1=K3
    v2f afrag[6];
#pragma unroll
    for (int kk = 0; kk < 6; ++kk) {
        int k0 = 4 * kk + 2 * half;
        int t0 = k0 - 2 * s;
        int t1 = t0 + 1;
        afrag[kk].x = (t0 >= 0 && t0 < 8) ? WFLIP[f * 8 + t0] : 0.0f;
        afrag[kk].y = (t1 >= 0 && t1 < 8) ? WFLIP[f * 8 + t1] : 0.0f;
    }

    const int Nin_tab[5]  = {16000, 8003, 4005, 2006, 1006};
    const int Kin_tab[5]  = {1, 2, 4, 8, 16};
    const int Nout_tab[5] = {8003, 4005, 2006, 1006, 506};

    float* srcb = lds;
    float* dstb = lds + BUF_STRIDE;

    __syncthreads();

    for (int lvl = 0; lvl < 5; ++lvl) {
        const int N    = Nin_tab[lvl];
        const int K    = Kin_tab[lvl];
        const int Nout = Nout_tab[lvl];
        const int ngroups = (Nout + 127) >> 7;          // 128 output positions per wave-task
        const int ntasks  = K * ngroups;

        for (int task = wave; task < ntasks; task += 8) {
            const int kc = task / ngroups;
            const int g  = task - kc * ngroups;
            const int jb = g << 7;                      // j_base
            const float* cin = srcb + kc * N;

            v8f acc = {};
            // B[k][n] = ext[2*j_base + 16n + k] = c_sym(2*j_base + 16n + k - 6)
            const int vb = 2 * jb + 16 * mrow - 6;
#pragma unroll
            for (int kk = 0; kk < 6; ++kk) {
                const int kb = 4 * kk + 2 * half;       // B rows: V0 = rows {0,2}+4kk, V1 = rows {1,3}+4kk
                v2f bfrag;
                bfrag.x = cin[symref(vb + kb,     N)];
                bfrag.y = cin[symref(vb + kb + 1, N)];
                acc = __builtin_amdgcn_wmma_f32_16x16x4_f32(
                    /*neg_a=*/false, afrag[kk],
                    /*neg_b=*/false, bfrag,
                    /*c_mod=*/(short)0, acc,
                    /*reuse_a=*/false, /*reuse_b=*/false);
            }

            // D layout: lane, VGPR v -> M = v + 8*half (= f*8 + s with s=v), N = lane&15
            // D[f*8+s][n] = y_f[j_base + 8n + s]
            float* cout = dstb + (2 * kc + half) * Nout;
#pragma unroll
            for (int v = 0; v < 8; ++v) {
                const int j = jb + 8 * mrow + v;
                if (j < Nout) cout[j] = acc[v];
            }
        }
        __syncthreads();
        float* t = srcb; srcb = dstb; dstb = t;
    }

    // ---- final coefficients: 32 channels x 506, packed flat at start of batch slab ----
    for (int i = tid; i < 32 * 506; i += 256) {
        out[(long)b * tot + i] = srcb[i];
    }
}

// ---------------------------------------------------------------------------
extern "C" void kernel_launch(void* const* d_in, const int* in_sizes, int n_in,
                              void* d_out, int out_size, void* d_ws, size_t ws_size,
                              hipStream_t stream) {
    const float* x = (const float*)d_in[0];
    float* out = (float*)d_out;

    // 1) zero-fill the whole output (padding dominates: ~262 MB, NT streaming stores)
    const long n = (long)out_size;
    const long n4 = (n + 3) / 4;
    const int  threads = 256;
    const long blocks = (n4 + threads - 1) / threads;
    fill_zero_kernel<<<dim3((unsigned)blocks), threads, 0, stream>>>(out, n);

    // 2) wavelet packet transform, one block per batch element
    const int tot = out_size / 8;   // per-batch output slab (2862*2862)
    wavelet_wmma_kernel<<<8, 256, 0, stream>>>(x, out, tot);
}